// AttentiveFP_46866683134577
// MI455X (gfx1250) — compile-verified
//
#include <hip/hip_runtime.h>

typedef __attribute__((ext_vector_type(2))) float v2f;
typedef __attribute__((ext_vector_type(8))) float v8f;

static constexpr int V  = 50000;
static constexpr int E  = 640000;
static constexpr int G  = 2500;
static constexpr int GF = 128;

// ---------------- activation helpers ----------------
template<int ACT> __device__ __forceinline__ float actf(float x){
  if (ACT == 1) return x > 0.f ? x : 0.01f * x;   // leaky relu
  if (ACT == 2) return x > 0.f ? x : 0.f;          // relu
  return x;
}

// ---------------- A-matrix loaders for WMMA GEMM ----------------
struct ALoadDense {
  const float* A; int K; const float* p;
  __device__ __forceinline__ void init(int r){ p = A + (size_t)r * K; }
  __device__ __forceinline__ float2 ld(int k) const { return *(const float2*)(p + k); }
};
// fused gather+concat: row e = [node_feats[src[e]] (64) , edge_feats[e] (16)], K=80
struct ALoadCtxEdge {
  const float* nf; const float* ef; const int* src; const float* pn; const float* pe;
  __device__ __forceinline__ void init(int e){
    pn = nf + (size_t)src[e] * 64;
    pe = ef + (size_t)e * 16 - 64;
  }
  __device__ __forceinline__ float2 ld(int k) const {
    return (k < 64) ? *(const float2*)(pn + k) : *(const float2*)(pe + k);
  }
};

// ---------------- WMMA fp32 GEMM: out[M,N] = act(A[M,K] @ W[N,K]^T + bias) ----------
// wave computes 16 rows x 64 cols; block = 4 waves = 64 rows. grid = (ceil(M/64), N/64)
// FULL: M is a multiple of 16 -> no per-lane row guards in the hot loop.
template<class AL, int ACT, int K, bool FULL>
__global__ void gemm_bias_act(AL al, int M, const float* __restrict__ W,
                              const float* __restrict__ bias,
                              float* __restrict__ out, int N){
  const int lane = threadIdx.x & 31;
  const int wave = threadIdx.x >> 5;
  const int r0 = (blockIdx.x * 4 + wave) * 16;
  if (r0 >= M) return;
  const int c0 = blockIdx.y * 64;
  const int m  = lane & 15;            // A: row-in-tile ; B: col-in-tile ; C: N index
  const int kh = (lane >> 4) << 1;     // K sub-offset 0 or 2
  const int row = r0 + m;
  const bool rok = FULL || (row < M);
  if (rok) al.init(row);
  v8f a0 = {0,0,0,0,0,0,0,0}, a1 = a0, a2 = a0, a3 = a0;
  const float* wp = W + (size_t)(c0 + m) * K + kh;
  #pragma unroll
  for (int k = 0; k < K; k += 4){
    float2 af = rok ? al.ld(k + kh) : make_float2(0.f, 0.f);
    v2f av; av.x = af.x; av.y = af.y;
    float2 b0 = *(const float2*)(wp + k);
    float2 b1 = *(const float2*)(wp + k + (size_t)16 * K);
    float2 b2 = *(const float2*)(wp + k + (size_t)32 * K);
    float2 b3 = *(const float2*)(wp + k + (size_t)48 * K);
    v2f bv;
    bv.x=b0.x; bv.y=b0.y; a0=__builtin_amdgcn_wmma_f32_16x16x4_f32(false,av,false,bv,(short)0,a0,false,false);
    bv.x=b1.x; bv.y=b1.y; a1=__builtin_amdgcn_wmma_f32_16x16x4_f32(false,av,false,bv,(short)0,a1,false,false);
    bv.x=b2.x; bv.y=b2.y; a2=__builtin_amdgcn_wmma_f32_16x16x4_f32(false,av,false,bv,(short)0,a2,false,false);
    bv.x=b3.x; bv.y=b3.y; a3=__builtin_amdgcn_wmma_f32_16x16x4_f32(false,av,false,bv,(short)0,a3,false,false);
  }
  const float bb0 = bias[c0 + m], bb1 = bias[c0 + 16 + m];
  const float bb2 = bias[c0 + 32 + m], bb3 = bias[c0 + 48 + m];
  const int rbase = r0 + ((lane >> 4) << 3);
  #pragma unroll
  for (int v = 0; v < 8; ++v){
    int rr = rbase + v;
    if (FULL || rr < M){
      size_t rs = (size_t)rr * N + c0 + m;
      out[rs]      = actf<ACT>(a0[v] + bb0);
      out[rs + 16] = actf<ACT>(a1[v] + bb1);
      out[rs + 32] = actf<ACT>(a2[v] + bb2);
      out[rs + 48] = actf<ACT>(a3[v] + bb3);
    }
  }
}

// ---- WMMA GEMM with fused segment-sum epilogue (K=128, M multiple of 16):
// out[map[r], :] += scale[r] * (A[r,:] @ W^T + bias)  (out has 128 cols)
__global__ void gemm_scatter_add(const float* __restrict__ A, int M,
                                 const float* __restrict__ W, const float* __restrict__ bias,
                                 const float* __restrict__ scale, const int* __restrict__ map,
                                 float* __restrict__ out){
  constexpr int K = 128;
  const int lane = threadIdx.x & 31;
  const int wave = threadIdx.x >> 5;
  const int r0 = (blockIdx.x * 4 + wave) * 16;
  if (r0 >= M) return;
  const int c0 = blockIdx.y * 64;
  const int m  = lane & 15;
  const int kh = (lane >> 4) << 1;
  const int row = r0 + m;
  const float* ap = A + (size_t)row * K;
  v8f a0 = {0,0,0,0,0,0,0,0}, a1 = a0, a2 = a0, a3 = a0;
  const float* wp = W + (size_t)(c0 + m) * K + kh;
  #pragma unroll
  for (int k = 0; k < K; k += 4){
    float2 af = *(const float2*)(ap + k + kh);
    v2f av; av.x = af.x; av.y = af.y;
    float2 b0 = *(const float2*)(wp + k);
    float2 b1 = *(const float2*)(wp + k + (size_t)16 * K);
    float2 b2 = *(const float2*)(wp + k + (size_t)32 * K);
    float2 b3 = *(const float2*)(wp + k + (size_t)48 * K);
    v2f bv;
    bv.x=b0.x; bv.y=b0.y; a0=__builtin_amdgcn_wmma_f32_16x16x4_f32(false,av,false,bv,(short)0,a0,false,false);
    bv.x=b1.x; bv.y=b1.y; a1=__builtin_amdgcn_wmma_f32_16x16x4_f32(false,av,false,bv,(short)0,a1,false,false);
    bv.x=b2.x; bv.y=b2.y; a2=__builtin_amdgcn_wmma_f32_16x16x4_f32(false,av,false,bv,(short)0,a2,false,false);
    bv.x=b3.x; bv.y=b3.y; a3=__builtin_amdgcn_wmma_f32_16x16x4_f32(false,av,false,bv,(short)0,a3,false,false);
  }
  const float bb0 = bias[c0 + m], bb1 = bias[c0 + 16 + m];
  const float bb2 = bias[c0 + 32 + m], bb3 = bias[c0 + 48 + m];
  const int rbase = r0 + ((lane >> 4) << 3);
  #pragma unroll
  for (int v = 0; v < 8; ++v){
    int rr = rbase + v;
    float s = scale[rr];
    size_t ob = (size_t)map[rr] * GF + c0 + m;
    atomicAdd(&out[ob],      s * (a0[v] + bb0));
    atomicAdd(&out[ob + 16], s * (a1[v] + bb1));
    atomicAdd(&out[ob + 32], s * (a2[v] + bb2));
    atomicAdd(&out[ob + 48], s * (a3[v] + bb3));
  }
}

// ---------------- elementwise / reduction kernels ----------------
__global__ void fill_f32(float* p, float v, size_t n){
  size_t i  = (size_t)blockIdx.x * blockDim.x + threadIdx.x;
  size_t st = (size_t)gridDim.x * blockDim.x;
  for (; i < n; i += st) p[i] = v;
}

__global__ void elu_inplace(float* p, size_t n){
  size_t i  = (size_t)blockIdx.x * blockDim.x + threadIdx.x;
  size_t st = (size_t)gridDim.x * blockDim.x;
  for (; i < n; i += st){ float x = p[i]; p[i] = x > 0.f ? x : (expf(x) - 1.f); }
}

__device__ __forceinline__ unsigned fkey(float f){
  unsigned b = __float_as_uint(f);
  return b ^ ((b >> 31) ? 0xFFFFFFFFu : 0x80000000u);
}
__device__ __forceinline__ float unkey(unsigned k){
  return __uint_as_float(k ^ ((k >> 31) ? 0x80000000u : 0xFFFFFFFFu));
}
__global__ void seg_max_k(const float* l, const int* seg, unsigned* key, int n){
  int i = blockIdx.x * blockDim.x + threadIdx.x;
  if (i < n) atomicMax(&key[seg[i]], fkey(l[i]));
}
__global__ void seg_exp_k(float* l, const int* seg, const unsigned* key, float* s, int n){
  int i = blockIdx.x * blockDim.x + threadIdx.x;
  if (i < n){
    float e = expf(l[i] - unkey(key[seg[i]]));
    l[i] = e;
    atomicAdd(&s[seg[i]], e);
  }
}
__global__ void seg_nrm_k(float* l, const int* seg, const float* s, int n){
  int i = blockIdx.x * blockDim.x + threadIdx.x;
  if (i < n) l[i] = l[i] / s[seg[i]];
}

// out[map[r], c] += (scale?scale[r]:1) * val[(vidx?vidx[r]:r), c]   (128 cols)
__global__ void scatter_add_rows(float* out, const int* __restrict__ map,
                                 const float* __restrict__ val, const int* __restrict__ vidx,
                                 const float* __restrict__ scale, int rows){
  size_t t = (size_t)blockIdx.x * blockDim.x + threadIdx.x;
  if (t >= (size_t)rows * GF) return;
  int r = (int)(t >> 7), c = (int)(t & 127);
  int sr = vidx ? vidx[r] : r;
  float v = val[(size_t)sr * GF + c];
  if (scale) v *= scale[r];
  atomicAdd(&out[(size_t)map[r] * GF + c], v);
}

// GRU gate math (torch GRUCell), gi/gh = [rows,384], h = [rows,128]
__global__ void gru_combine(const float* __restrict__ gi, const float* __restrict__ gh,
                            const float* __restrict__ hold, float* __restrict__ hout,
                            int rows, int relu_out){
  size_t t = (size_t)blockIdx.x * blockDim.x + threadIdx.x;
  if (t >= (size_t)rows * GF) return;
  int r = (int)(t >> 7), c = (int)(t & 127);
  const float* gir = gi + (size_t)r * 384;
  const float* ghr = gh + (size_t)r * 384;
  float ir = gir[c], iz = gir[128 + c], in_ = gir[256 + c];
  float hr = ghr[c], hz = ghr[128 + c], hn  = ghr[256 + c];
  float rg = 1.f / (1.f + expf(-(ir + hr)));
  float z  = 1.f / (1.f + expf(-(iz + hz)));
  float nn = tanhf(in_ + rg * hn);
  float h  = hold[(size_t)r * GF + c];
  float o  = (1.f - z) * nn + z * h;
  if (relu_out) o = o > 0.f ? o : 0.f;
  hout[(size_t)r * GF + c] = o;
}

// out[i] = lrelu(w[0:128].X1[idx1?idx1[i]:i] (opt relu) + w[128:256].X2[idx2?idx2[i]:i] + b)
// one wave (32 lanes) per item
__global__ void pair_logit(float* __restrict__ out, const float* __restrict__ w,
                           const float* __restrict__ bp,
                           const float* __restrict__ X1, const int* __restrict__ idx1, int relu1,
                           const float* __restrict__ X2, const int* __restrict__ idx2, int n){
  int wave = threadIdx.x >> 5, lane = threadIdx.x & 31;
  int i = blockIdx.x * 4 + wave;
  if (i >= n) return;
  int i1 = idx1 ? idx1[i] : i;
  int i2 = idx2 ? idx2[i] : i;
  const float* x1 = X1 + (size_t)i1 * GF;
  const float* x2 = X2 + (size_t)i2 * GF;
  float s = 0.f;
  #pragma unroll
  for (int j = 0; j < 4; ++j){
    int c = lane + 32 * j;
    float v1 = x1[c];
    if (relu1) v1 = v1 > 0.f ? v1 : 0.f;
    s += w[c] * v1 + w[128 + c] * x2[c];
  }
  for (int off = 16; off; off >>= 1) s += __shfl_xor(s, off, 32);
  if (lane == 0){ float z = s + bp[0]; out[i] = z > 0.f ? z : 0.01f * z; }
}

// out[i] = w . X[i,:] + b   (128 cols, wave per row)
__global__ void rowdot(float* __restrict__ out, const float* __restrict__ w,
                       const float* __restrict__ bp, const float* __restrict__ X, int n){
  int wave = threadIdx.x >> 5, lane = threadIdx.x & 31;
  int i = blockIdx.x * 4 + wave;
  if (i >= n) return;
  float s = 0.f;
  #pragma unroll
  for (int j = 0; j < 4; ++j){ int c = lane + 32 * j; s += w[c] * X[(size_t)i * GF + c]; }
  for (int off = 16; off; off >>= 1) s += __shfl_xor(s, off, 32);
  if (lane == 0) out[i] = s + bp[0];
}

// ---------------- host orchestration ----------------
struct P {
  const float *ctx_node_w, *ctx_node_b, *e1_w, *e1_b, *e2_w, *e2_b, *et_w, *et_b;
  const float *ctx_wih, *ctx_whh, *ctx_bih, *ctx_bhh;
  const float *L_edge_w[2], *L_edge_b[2], *L_node_w[2], *L_node_b[2];
  const float *L_wih[2], *L_whh[2], *L_bih[2], *L_bhh[2];
  const float *R_log_w[2], *R_log_b[2], *R_node_w[2], *R_node_b[2];
  const float *R_wih[2], *R_whh[2], *R_bih[2], *R_bhh[2];
  const float *pred_w, *pred_b;
};

extern "C" void kernel_launch(void* const* d_in, const int* in_sizes, int n_in,
                              void* d_out, int out_size, void* d_ws, size_t ws_size,
                              hipStream_t stream){
  if (n_in < 51) return;
  const float* node_feats = (const float*)d_in[0];
  const float* edge_feats = (const float*)d_in[1];
  const int*   src        = (const int*)d_in[2];
  const int*   dst        = (const int*)d_in[3];
  const int*   n2g        = (const int*)d_in[4];

  auto F = [&](int i){ return (const float*)d_in[i]; };
  P p;
  if (in_sizes[5] == 128){
    // jax tree_flatten order (dict keys sorted)
    p.e1_b = F(5);  p.e1_w = F(6);  p.e2_b = F(7);  p.e2_w = F(8);
    p.et_b = F(9);  p.et_w = F(10);
    p.ctx_bhh = F(11); p.ctx_bih = F(12); p.ctx_whh = F(13); p.ctx_wih = F(14);
    p.ctx_node_b = F(15); p.ctx_node_w = F(16);
    for (int l = 0; l < 2; ++l){
      int b = 17 + 8 * l;
      p.L_edge_b[l]=F(b+0); p.L_edge_w[l]=F(b+1);
      p.L_bhh[l]=F(b+2); p.L_bih[l]=F(b+3); p.L_whh[l]=F(b+4); p.L_wih[l]=F(b+5);
      p.L_node_b[l]=F(b+6); p.L_node_w[l]=F(b+7);
    }
    p.pred_b = F(33); p.pred_w = F(34);
    for (int r = 0; r < 2; ++r){
      int b = 35 + 8 * r;
      p.R_bhh[r]=F(b+0); p.R_bih[r]=F(b+1); p.R_whh[r]=F(b+2); p.R_wih[r]=F(b+3);
      p.R_log_b[r]=F(b+4); p.R_log_w[r]=F(b+5);
      p.R_node_b[r]=F(b+6); p.R_node_w[r]=F(b+7);
    }
  } else {
    // python dict insertion order
    p.ctx_node_w = F(5);  p.ctx_node_b = F(6);
    p.e1_w = F(7);  p.e1_b = F(8);  p.e2_w = F(9);  p.e2_b = F(10);
    p.et_w = F(11); p.et_b = F(12);
    p.ctx_wih = F(13); p.ctx_whh = F(14); p.ctx_bih = F(15); p.ctx_bhh = F(16);
    for (int l = 0; l < 2; ++l){
      int b = 17 + 8 * l;
      p.L_edge_w[l]=F(b+0); p.L_edge_b[l]=F(b+1);
      p.L_node_w[l]=F(b+2); p.L_node_b[l]=F(b+3);
      p.L_wih[l]=F(b+4); p.L_whh[l]=F(b+5); p.L_bih[l]=F(b+6); p.L_bhh[l]=F(b+7);
    }
    for (int r = 0; r < 2; ++r){
      int b = 33 + 8 * r;
      p.R_log_w[r]=F(b+0); p.R_log_b[r]=F(b+1);
      p.R_node_w[r]=F(b+2); p.R_node_b[r]=F(b+3);
      p.R_wih[r]=F(b+4); p.R_whh[r]=F(b+5); p.R_bih[r]=F(b+6); p.R_bhh[r]=F(b+7);
    }
    p.pred_w = F(49); p.pred_b = F(50);
  }

  // ---- workspace layout (floats) ----
  float* ws = (float*)d_ws;
  size_t off = 0;
  auto alloc = [&](size_t n){ float* q = ws + off; off += n; return q; };
  float*    hv   = alloc((size_t)V * GF);
  float*    h    = alloc((size_t)V * GF);
  float*    he1  = alloc((size_t)E * GF);
  float*    elog = alloc(E);
  unsigned* ikey = (unsigned*)alloc(V);   // adjacent to ssum: zero both in one fill
  float*    ssum = alloc(V);
  float*    ctx  = alloc((size_t)V * GF);
  float*    hp   = alloc((size_t)V * GF);
  float*    gf   = alloc((size_t)G * GF);
  float*    ctxg = alloc((size_t)G * GF);
  float*    gi   = alloc((size_t)V * 384);
  float*    gh   = alloc((size_t)V * 384);
  if (off * sizeof(float) > ws_size) return;
  (void)out_size;

  const int  TB  = 256;
  const dim3 blk(128);
  auto nb = [&](size_t n){ return dim3((unsigned)((n + TB - 1) / TB)); };

  auto softmax = [&](float* logit, const int* seg, int n){
    fill_f32<<<1024, TB, 0, stream>>>((float*)ikey, 0.f, (size_t)2 * V);
    seg_max_k<<<nb(n), TB, 0, stream>>>(logit, seg, ikey, n);
    seg_exp_k<<<nb(n), TB, 0, stream>>>(logit, seg, ikey, ssum, n);
    seg_nrm_k<<<nb(n), TB, 0, stream>>>(logit, seg, ssum, n);
  };

  const dim3 gV2((V + 63) / 64, 2), gV6((V + 63) / 64, 6);
  const dim3 gE2((E + 63) / 64, 2);
  const dim3 gG2((G + 63) / 64, 2), gG6((G + 63) / 64, 6);

  // ================= GetContext =================
  { ALoadDense al{node_feats, 64, nullptr};
    gemm_bias_act<ALoadDense, 1, 64, true><<<gV2, blk, 0, stream>>>(al, V, p.ctx_node_w, p.ctx_node_b, hv, GF); }
  { ALoadCtxEdge al{node_feats, edge_feats, src, nullptr, nullptr};
    gemm_bias_act<ALoadCtxEdge, 1, 80, true><<<gE2, blk, 0, stream>>>(al, E, p.e1_w, p.e1_b, he1, GF); }
  pair_logit<<<dim3(E / 4), blk, 0, stream>>>(elog, p.e2_w, p.e2_b, hv, dst, 0, he1, nullptr, E);
  softmax(elog, dst, E);
  fill_f32<<<4096, TB, 0, stream>>>(ctx, 0.f, (size_t)V * GF);
  gemm_scatter_add<<<gE2, blk, 0, stream>>>(he1, E, p.et_w, p.et_b, elog, dst, ctx);
  elu_inplace<<<4096, TB, 0, stream>>>(ctx, (size_t)V * GF);
  { ALoadDense al{ctx, GF, nullptr};
    gemm_bias_act<ALoadDense, 0, 128, true><<<gV6, blk, 0, stream>>>(al, V, p.ctx_wih, p.ctx_bih, gi, 384); }
  { ALoadDense al{hv, GF, nullptr};
    gemm_bias_act<ALoadDense, 0, 128, true><<<gV6, blk, 0, stream>>>(al, V, p.ctx_whh, p.ctx_bhh, gh, 384); }
  gru_combine<<<nb((size_t)V * GF), TB, 0, stream>>>(gi, gh, hv, h, V, 1);

  // ================= GNN layers =================
  for (int l = 0; l < 2; ++l){
    pair_logit<<<dim3(E / 4), blk, 0, stream>>>(elog, p.L_edge_w[l], p.L_edge_b[l], h, dst, 0, h, src, E);
    softmax(elog, dst, E);
    { ALoadDense al{h, GF, nullptr};
      gemm_bias_act<ALoadDense, 0, 128, true><<<gV2, blk, 0, stream>>>(al, V, p.L_node_w[l], p.L_node_b[l], hp, GF); }
    fill_f32<<<4096, TB, 0, stream>>>(ctx, 0.f, (size_t)V * GF);
    scatter_add_rows<<<nb((size_t)E * GF), TB, 0, stream>>>(ctx, dst, hp, src, elog, E);
    elu_inplace<<<4096, TB, 0, stream>>>(ctx, (size_t)V * GF);
    { ALoadDense al{ctx, GF, nullptr};
      gemm_bias_act<ALoadDense, 0, 128, true><<<gV6, blk, 0, stream>>>(al, V, p.L_wih[l], p.L_bih[l], gi, 384); }
    { ALoadDense al{h, GF, nullptr};
      gemm_bias_act<ALoadDense, 0, 128, true><<<gV6, blk, 0, stream>>>(al, V, p.L_whh[l], p.L_bhh[l], gh, 384); }
    gru_combine<<<nb((size_t)V * GF), TB, 0, stream>>>(gi, gh, h, h, V, 1);
  }

  // ================= Readout =================
  fill_f32<<<64, TB, 0, stream>>>(gf, 0.f, (size_t)G * GF);
  scatter_add_rows<<<nb((size_t)V * GF), TB, 0, stream>>>(gf, n2g, h, nullptr, nullptr, V);
  for (int r = 0; r < 2; ++r){
    pair_logit<<<dim3((V + 3) / 4), blk, 0, stream>>>(elog, p.R_log_w[r], p.R_log_b[r], gf, n2g, 1, h, nullptr, V);
    softmax(elog, n2g, V);
    { ALoadDense al{h, GF, nullptr};
      gemm_bias_act<ALoadDense, 0, 128, true><<<gV2, blk, 0, stream>>>(al, V, p.R_node_w[r], p.R_node_b[r], hp, GF); }
    fill_f32<<<64, TB, 0, stream>>>(ctxg, 0.f, (size_t)G * GF);
    scatter_add_rows<<<nb((size_t)V * GF), TB, 0, stream>>>(ctxg, n2g, hp, nullptr, elog, V);
    elu_inplace<<<64, TB, 0, stream>>>(ctxg, (size_t)G * GF);
    { ALoadDense al{ctxg, GF, nullptr};
      gemm_bias_act<ALoadDense, 0, 128, false><<<gG6, blk, 0, stream>>>(al, G, p.R_wih[r], p.R_bih[r], gi, 384); }
    { ALoadDense al{gf, GF, nullptr};
      gemm_bias_act<ALoadDense, 0, 128, false><<<gG6, blk, 0, stream>>>(al, G, p.R_whh[r], p.R_bhh[r], gh, 384); }
    gru_combine<<<nb((size_t)G * GF), TB, 0, stream>>>(gi, gh, gf, gf, G, 0);
  }
  rowdot<<<dim3((G + 3) / 4), blk, 0, stream>>>((float*)d_out, p.pred_w, p.pred_b, gf, G);
}